// TensorSplatter_31817117728922
// MI455X (gfx1250) — compile-verified
//
#include <hip/hip_runtime.h>
#include <hip/hip_bf16.h>
#include <math.h>

#define Q_N 8192
#define A_N 2048
#define P_N (A_N / 2)          // atom pairs
#define D_N 64
#define R2ROWS 20              // K-slots per atom pair (2 x (9 channels + 1 pad))
#define R2STRIDE (R2ROWS * 16) // 320 floats per pair

typedef __attribute__((ext_vector_type(2))) float v2f;
typedef __attribute__((ext_vector_type(8))) float v8f;

// ---------------------------------------------------------------------------
// Kernel A: per-atom readout coefficients -> half-interleaved B-operand matrix
// R2[pair][20][16], K-slot map (g = K-group 0..4, h = atom parity):
//   slot 4g + 2h + 0 = atom(2*pair+h) channel 2g
//   slot 4g + 2h + 1 = atom(2*pair+h) channel 2g+1   (ch 9 == zero pad)
// This guarantees lanes 0-15 of the WMMA A-operand only need atom0's channels
// and lanes 16-31 only atom1's -> no cross-lane traffic in the hot loop.
// Channel meaning (per atom):
//   ch=0      "w"      -> [1, c0e, c0o, dip(3), biv(3), bE, bO, 0..]
//   ch=1..3   "w*d_j"  -> [0,0,0, Q[:,j](3), Qo[:,j](3), av1o[j], av1e[j], 0..]
//   ch=4..8   "w*y2_m" -> [0 x9, at2e[m], at2o[m], 0..]
// ---------------------------------------------------------------------------
__global__ void __launch_bounds__(256)
atom_precompute_kernel(const float* __restrict__ even_scalar,
                       const float* __restrict__ odd_scalar,
                       const float* __restrict__ odd_vector,
                       const float* __restrict__ even_vector,
                       const float* __restrict__ even_tensor,
                       const float* __restrict__ odd_tensor,
                       const float* __restrict__ W,   // [10,64]
                       const float* __restrict__ b,   // [10]
                       float* __restrict__ R2)        // [P_N,20,16]
{
    int a = blockIdx.x * blockDim.x + threadIdx.x;
    if (a >= A_N) return;

    const float* esA = even_scalar + a * D_N;
    const float* osA = odd_scalar  + a * D_N;
    const float* ovA = odd_vector  + a * D_N * 3;
    const float* evA = even_vector + a * D_N * 3;
    const float* etA = even_tensor + a * D_N * 5;
    const float* otA = odd_tensor  + a * D_N * 5;

    float c0e = 0.f, c0o = 0.f;
    float dip[3]  = {0.f, 0.f, 0.f}, biv[3]  = {0.f, 0.f, 0.f};
    float av1o[3] = {0.f, 0.f, 0.f}, av1e[3] = {0.f, 0.f, 0.f};
    float qc[5]  = {0.f, 0.f, 0.f, 0.f, 0.f};
    float oqc[5] = {0.f, 0.f, 0.f, 0.f, 0.f};
    float at2e[5] = {0.f, 0.f, 0.f, 0.f, 0.f};
    float at2o[5] = {0.f, 0.f, 0.f, 0.f, 0.f};

    for (int d = 0; d < D_N; ++d) {
        float w0 = W[0 * D_N + d], w1 = W[1 * D_N + d];
        float w2 = W[2 * D_N + d], w3 = W[3 * D_N + d];
        float w4 = W[4 * D_N + d], w5 = W[5 * D_N + d];
        float w6 = W[6 * D_N + d], w7 = W[7 * D_N + d];
        float w8 = W[8 * D_N + d], w9 = W[9 * D_N + d];
        c0e = fmaf(esA[d], w0, c0e);
        c0o = fmaf(osA[d], w1, c0o);
#pragma unroll
        for (int i = 0; i < 3; ++i) {
            float o_ = ovA[d * 3 + i];
            float e_ = evA[d * 3 + i];
            dip[i]  = fmaf(o_, w2, dip[i]);
            av1o[i] = fmaf(o_, w6, av1o[i]);
            biv[i]  = fmaf(e_, w3, biv[i]);
            av1e[i] = fmaf(e_, w7, av1e[i]);
        }
#pragma unroll
        for (int k = 0; k < 5; ++k) {
            float te = etA[d * 5 + k];
            float to = otA[d * 5 + k];
            qc[k]   = fmaf(te, w4, qc[k]);
            at2e[k] = fmaf(te, w8, at2e[k]);
            oqc[k]  = fmaf(to, w5, oqc[k]);
            at2o[k] = fmaf(to, w9, at2o[k]);
        }
    }
    // biases
    c0e += b[0];
    c0o += b[1];
#pragma unroll
    for (int i = 0; i < 3; ++i) { dip[i] += b[2]; biv[i] += b[3]; }
#pragma unroll
    for (int k = 0; k < 5; ++k) { qc[k] += b[4]; oqc[k] += b[5]; }
    float bE = b[6] + b[8];   // angular_1o + angular_2e biases (even path)
    float bO = b[7] + b[9];   // angular_1e + angular_2o biases (odd path)

    // reconstruct symmetric traceless quadrupoles
    const float SQ2   = 1.41421356237309515f;
    const float ISQ2  = 0.70710678118654752f;
    const float SQ2_3 = 0.816496580927726f;
    float Qm[3][3], Qo[3][3];
    {
        float zz = qc[1] * SQ2_3;
        float xx = 0.5f * (qc[0] * SQ2 - zz);
        float yy = 0.5f * (-qc[0] * SQ2 - zz);
        float xy = qc[2] * ISQ2, xz = qc[3] * ISQ2, yz = qc[4] * ISQ2;
        Qm[0][0] = xx; Qm[0][1] = xy; Qm[0][2] = xz;
        Qm[1][0] = xy; Qm[1][1] = yy; Qm[1][2] = yz;
        Qm[2][0] = xz; Qm[2][1] = yz; Qm[2][2] = zz;
    }
    {
        float zz = oqc[1] * SQ2_3;
        float xx = 0.5f * (oqc[0] * SQ2 - zz);
        float yy = 0.5f * (-oqc[0] * SQ2 - zz);
        float xy = oqc[2] * ISQ2, xz = oqc[3] * ISQ2, yz = oqc[4] * ISQ2;
        Qo[0][0] = xx; Qo[0][1] = xy; Qo[0][2] = xz;
        Qo[1][0] = xy; Qo[1][1] = yy; Qo[1][2] = yz;
        Qo[2][0] = xz; Qo[2][1] = yz; Qo[2][2] = zz;
    }

    int pair = a >> 1;
    int half = a & 1;
    float* Rp = R2 + pair * R2STRIDE;
    // slot offset (in floats) for this atom's channel ch
    auto slotOff = [&](int ch) { return (4 * (ch >> 1) + 2 * half + (ch & 1)) * 16; };
    float row[16];

    // ch = 0 : "w" channel
#pragma unroll
    for (int n = 0; n < 16; ++n) row[n] = 0.f;
    row[0] = 1.0f;
    row[1] = c0e; row[2] = c0o;
    row[3] = dip[0]; row[4] = dip[1]; row[5] = dip[2];
    row[6] = biv[0]; row[7] = biv[1]; row[8] = biv[2];
    row[9] = bE; row[10] = bO;
    {
        float* dst = Rp + slotOff(0);
#pragma unroll
        for (int n = 0; n < 16; ++n) dst[n] = row[n];
    }

    // ch = 1..3 : "w*d_j" channels
#pragma unroll
    for (int j = 0; j < 3; ++j) {
#pragma unroll
        for (int n = 0; n < 16; ++n) row[n] = 0.f;
        row[3] = Qm[0][j]; row[4] = Qm[1][j]; row[5] = Qm[2][j];
        row[6] = Qo[0][j]; row[7] = Qo[1][j]; row[8] = Qo[2][j];
        row[9] = av1o[j];  row[10] = av1e[j];
        float* dst = Rp + slotOff(1 + j);
#pragma unroll
        for (int n = 0; n < 16; ++n) dst[n] = row[n];
    }

    // ch = 4..8 : "w*y2_m" channels
#pragma unroll
    for (int m = 0; m < 5; ++m) {
#pragma unroll
        for (int n = 0; n < 16; ++n) row[n] = 0.f;
        row[9] = at2e[m]; row[10] = at2o[m];
        float* dst = Rp + slotOff(4 + m);
#pragma unroll
        for (int n = 0; n < 16; ++n) dst[n] = row[n];
    }

    // ch = 9 : this atom's zero-pad slot
    {
        float* dst = Rp + slotOff(9);
#pragma unroll
        for (int n = 0; n < 16; ++n) dst[n] = 0.f;
    }
}

// ---------------------------------------------------------------------------
// Kernel B: main Q x A accumulation via V_WMMA_F32_16X16X4_F32.
// One wave32 handles 16 queries (M=16). Per iteration the wave consumes one
// ATOM PAIR: lanes 0-15 compute geometry for atom 2it, lanes 16-31 for atom
// 2it+1. Thanks to the half-interleaved K-slot packing, each lane feeds the
// A-operand exclusively from its own atom's channels: no cross-lane ops.
//   A-operand (16x4 f32): lanes 0-15 supply K=4g,4g+1 (atom0 ch 2g,2g+1);
//                         lanes 16-31 supply K=4g+2,4g+3 (atom1 ch 2g,2g+1).
// Direction: dist=sqrt(max(d2,1e-16)) >= 1e-8, so the max(dist,1e-8) clamp in
// the reference is a no-op and rel/dist_c gated by dist>1e-6 is exactly
// rel * rsq(d2) gated by d2>1e-12 -> single v_rsq_f32 on the TRANS pipe.
// ---------------------------------------------------------------------------
__global__ void __launch_bounds__(256)
splat_wmma_kernel(const float* __restrict__ query_coords,
                  const float* __restrict__ atom_coords,
                  const float* __restrict__ alpha,
                  const float* __restrict__ R2,
                  float* __restrict__ tile_out)   // [Q,16] raw accumulators
{
    __shared__ float4 satom[A_N];   // 32 KB: xyz + alpha
    for (int i = threadIdx.x; i < A_N; i += blockDim.x) {
        satom[i] = make_float4(atom_coords[3 * i + 0],
                               atom_coords[3 * i + 1],
                               atom_coords[3 * i + 2],
                               alpha[i]);
    }
    __syncthreads();

    const int lane  = threadIdx.x & 31;
    const int wave  = blockIdx.x * (blockDim.x >> 5) + (threadIdx.x >> 5);
    const int qbase = wave * 16;
    const int nsel  = lane & 15;
    const bool hi   = lane >= 16;
    const int q     = qbase + nsel;

    const float qx = query_coords[3 * q + 0];
    const float qy = query_coords[3 * q + 1];
    const float qz = query_coords[3 * q + 2];

    const float SQ2  = 1.41421356237309515f;
    const float ISQ2 = 0.70710678118654752f;
    const float ISQ6 = 0.40824829046386302f;

    v8f acc = {0.f, 0.f, 0.f, 0.f, 0.f, 0.f, 0.f, 0.f};

    for (int it = 0; it < P_N; ++it) {
        // prefetch B-operand a few pairs ahead (L2-resident stream)
        __builtin_prefetch(R2 + (it + 4) * R2STRIDE + nsel + (hi ? 32 : 0), 0, 1);

        // this lane's atom: low half -> 2it, high half -> 2it+1
        float4 at = satom[2 * it + (hi ? 1 : 0)];
        float rx = qx - at.x, ry = qy - at.y, rz = qz - at.z;
        float d2 = fmaf(rx, rx, fmaf(ry, ry, rz * rz));
        float w  = __expf(-at.w * d2);
        // direction = rel * rsq(d2), zeroed for near-coincident points
        float inv = (d2 > 1e-12f) ? __builtin_amdgcn_rsqf(d2) : 0.0f;
        float dx = rx * inv, dy = ry * inv, dz = rz * inv;
        float xx = dx * dx, yy = dy * dy, zz = dz * dz;
        // channels of THIS lane's atom: c0=w, c1..3 = w*d, c4..8 = w*y2
        float c0 = w;
        float c1 = w * dx, c2 = w * dy, c3 = w * dz;
        float c4 = w * ((xx - yy) * ISQ2);
        float c5 = w * ((-xx - yy + 2.f * zz) * ISQ6);
        float c6 = w * (SQ2 * dx * dy);
        float c7 = w * (SQ2 * dx * dz);
        float c8 = w * (SQ2 * dy * dz);

        // B-operand rows for this lane: K = 4g + (hi?2:0) and +1
        const float* rb = R2 + it * R2STRIDE + (hi ? 32 : 0) + nsel;

        v2f Aop, Bop;
        // g0: this atom's ch0, ch1
        Aop.x = c0; Aop.y = c1;
        Bop.x = rb[0];   Bop.y = rb[16];
        acc = __builtin_amdgcn_wmma_f32_16x16x4_f32(false, Aop, false, Bop,
                                                    (short)0, acc, false, false);
        // g1: ch2, ch3
        Aop.x = c2; Aop.y = c3;
        Bop.x = rb[64];  Bop.y = rb[80];
        acc = __builtin_amdgcn_wmma_f32_16x16x4_f32(false, Aop, false, Bop,
                                                    (short)0, acc, false, false);
        // g2: ch4, ch5
        Aop.x = c4; Aop.y = c5;
        Bop.x = rb[128]; Bop.y = rb[144];
        acc = __builtin_amdgcn_wmma_f32_16x16x4_f32(false, Aop, false, Bop,
                                                    (short)0, acc, false, false);
        // g3: ch6, ch7
        Aop.x = c6; Aop.y = c7;
        Bop.x = rb[192]; Bop.y = rb[208];
        acc = __builtin_amdgcn_wmma_f32_16x16x4_f32(false, Aop, false, Bop,
                                                    (short)0, acc, false, false);
        // g4: ch8, zero pad
        Aop.x = c8; Aop.y = 0.f;
        Bop.x = rb[256]; Bop.y = rb[272];
        acc = __builtin_amdgcn_wmma_f32_16x16x4_f32(false, Aop, false, Bop,
                                                    (short)0, acc, false, false);
    }

    // C/D layout: VGPR r -> (M=r, N=lane) for lanes 0..15, (M=8+r, N=lane-16) else
#pragma unroll
    for (int r = 0; r < 8; ++r) {
        int rowM = hi ? (8 + r) : r;
        tile_out[(qbase + rowM) * 16 + nsel] = acc[r];
    }
}

// ---------------------------------------------------------------------------
// Kernel C: normalize by weight sum, assemble outputs
// d_out = [ total[Q] | even[Q] | odd[Q] | multivector[Q,16] ]
// ---------------------------------------------------------------------------
__global__ void __launch_bounds__(256)
finalize_kernel(const float* __restrict__ tile, float* __restrict__ out)
{
    int q = blockIdx.x * blockDim.x + threadIdx.x;
    if (q >= Q_N) return;
    const float* t = tile + q * 16;
    float wsum = t[0];
    float invw = 1.0f / fmaxf(wsum, 1e-8f);
    float scalar = t[1] * invw;
    float pseudo = t[2] * invw;
    float vx = t[3] * invw, vy = t[4] * invw, vz = t[5] * invw;
    float bx = t[6] * invw, by = t[7] * invw, bz = t[8] * invw;
    float evAng = t[9] * invw, odAng = t[10] * invw;

    float evenC = scalar + evAng;
    float oddC  = pseudo + odAng;
    float vn = sqrtf(fmaf(vx, vx, fmaf(vy, vy, vz * vz)));
    float bn = sqrtf(fmaf(bx, bx, fmaf(by, by, bz * bz)));
    float total = evenC + oddC + 0.1f * vn + 0.05f * bn;

    out[q]            = total;
    out[Q_N + q]      = evenC;
    out[2 * Q_N + q]  = oddC;
    float* mv = out + 3 * Q_N + q * 16;
#pragma unroll
    for (int j = 0; j < 16; ++j) mv[j] = 0.f;
    mv[0]  = scalar;
    mv[1]  = vx; mv[2] = vy; mv[3] = vz;
    mv[5]  = bz; mv[6] = -by; mv[7] = bx;
    mv[11] = pseudo;
}

extern "C" void kernel_launch(void* const* d_in, const int* in_sizes, int n_in,
                              void* d_out, int out_size, void* d_ws, size_t ws_size,
                              hipStream_t stream) {
    const float* query_coords = (const float*)d_in[0];
    const float* atom_coords  = (const float*)d_in[1];
    const float* alpha        = (const float*)d_in[2];
    const float* even_scalar  = (const float*)d_in[3];
    const float* odd_scalar   = (const float*)d_in[4];
    const float* odd_vector   = (const float*)d_in[5];
    const float* even_vector  = (const float*)d_in[6];
    const float* even_tensor  = (const float*)d_in[7];
    const float* odd_tensor   = (const float*)d_in[8];
    const float* W_readout    = (const float*)d_in[9];
    const float* b_readout    = (const float*)d_in[10];
    float* out = (float*)d_out;

    float* R2   = (float*)d_ws;                  // P_N * 320 floats = 1.25 MB
    float* tile = R2 + (size_t)P_N * R2STRIDE;   // Q_N * 16 floats  = 0.5 MB

    // Stage 1: per-atom B-operand matrices (half-interleaved pair packing)
    atom_precompute_kernel<<<A_N / 256, 256, 0, stream>>>(
        even_scalar, odd_scalar, odd_vector, even_vector,
        even_tensor, odd_tensor, W_readout, b_readout, R2);

    // Stage 2: Q x A WMMA accumulation (512 waves of 16 queries each)
    splat_wmma_kernel<<<Q_N / (16 * 8), 256, 0, stream>>>(
        query_coords, atom_coords, alpha, R2, tile);

    // Stage 3: normalization + output assembly
    finalize_kernel<<<Q_N / 256, 256, 0, stream>>>(tile, out);
}